// DiffFDN_40269613367801
// MI455X (gfx1250) — compile-verified
//
#include <hip/hip_runtime.h>
#include <math.h>

// DiffFDN frequency response on gfx1250.
//
// H[f,:] = C ∘ (diag(z_f^m) - V·diag(g))^{-1} B,  V = expm(skew(A)), z_f = e^{iθ_f}
//
// Factor M = D(I - W), W = D^{-1}VG, ||W||2 <= 0.9999^593 ≈ 0.942  =>
// Neumann series x = Σ_k W^k (D^{-1}B) converges; ~320 terms give fp32 accuracy.
// W y = Z ∘ (VG·y) with Z[n,f] = e^{-i m_n θ_f}: a real 16x16 matmul against a
// complex 16-wide frequency batch -> v_wmma_f32_16x16x4_f32 chains per wave.

#define FDN_N    16
#define FDN_NUM  131072
#define NITER    320

typedef __attribute__((ext_vector_type(2))) float v2f;
typedef __attribute__((ext_vector_type(8))) float v8f;

__constant__ float c_delays[FDN_N] = {
    593.f, 743.f, 929.f, 1153.f, 1399.f, 1699.f, 1999.f, 2311.f,
    2671.f, 3089.f, 3539.f, 4051.f, 4603.f, 5209.f, 5857.f, 6553.f};

#define GAIN_PER_SAMPLE 0.9999f

// lane <-> lane^16 half-wave swap (wave32): ds_swizzle SWAPX16 pattern
__device__ __forceinline__ float swap_half(float x) {
    return __int_as_float(
        __builtin_amdgcn_ds_swizzle(__float_as_int(x), 0x401F));
}

// --------------------------------------------------------------------------
// Kernel 0: VG = expm(triu(A,1) - triu(A,1)^T) * diag(0.9999^m)
// One 256-thread block; thread (i,j) owns element (i,j). Scaling & squaring:
// X = K/2^8, R = Σ_{t<=10} X^t/t!, then square 8x. Error ~ (0.03)^11/11!.
// --------------------------------------------------------------------------
__global__ void __launch_bounds__(256) fdn_expm_kernel(const float* __restrict__ A,
                                                       float* __restrict__ VG) {
    __shared__ float X[16][16], T[16][16], R[16][16];
    const int i = threadIdx.x >> 4;
    const int j = threadIdx.x & 15;

    float a_ij = (j > i) ? A[i * 16 + j] : 0.f;  // triu(A,1)
    float a_ji = (i > j) ? A[j * 16 + i] : 0.f;  // triu(A,1)^T
    X[i][j] = (a_ij - a_ji) * (1.0f / 256.0f);
    T[i][j] = (i == j) ? 1.f : 0.f;
    R[i][j] = (i == j) ? 1.f : 0.f;
    __syncthreads();

    for (int t = 1; t <= 10; ++t) {   // Taylor: T <- X*T/t, R += T
        float acc = 0.f;
#pragma unroll
        for (int k = 0; k < 16; ++k) acc += X[i][k] * T[k][j];
        acc *= 1.0f / (float)t;
        __syncthreads();
        T[i][j] = acc;
        R[i][j] += acc;
        __syncthreads();
    }
    for (int s = 0; s < 8; ++s) {     // undo scaling: R <- R*R, 8 times
        float acc = 0.f;
#pragma unroll
        for (int k = 0; k < 16; ++k) acc += R[i][k] * R[k][j];
        __syncthreads();
        R[i][j] = acc;
        __syncthreads();
    }
    VG[i * 16 + j] = R[i][j] * powf(GAIN_PER_SAMPLE, c_delays[j]);
}

// D-layout (v8f accumulator) -> B-operand layout BL[chunk kk][reg j].
// D: vgpr r = row r (lanes 0-15) / row 8+r (lanes 16-31), lane = column.
// B chunk kk, reg j: row 4kk+j (lanes 0-15) / row 4kk+j+2 (lanes 16-31).
__device__ __forceinline__ void d_to_b(const v8f& D, float BL[4][2], int h) {
#pragma unroll
    for (int kk = 0; kk < 2; ++kk)
#pragma unroll
        for (int j = 0; j < 2; ++j) {
            float a = D[4 * kk + j];                 // rows 0..7: resident half
            float b = swap_half(D[4 * kk + j + 2]);  // partner lane's copy
            BL[kk][j] = h ? b : a;
        }
#pragma unroll
    for (int kk = 2; kk < 4; ++kk)
#pragma unroll
        for (int j = 0; j < 2; ++j) {
            float a = swap_half(D[4 * (kk - 2) + j]); // rows 8..15 live in upper half
            float b = D[4 * (kk - 2) + j + 2];
            BL[kk][j] = h ? b : a;
        }
}

// --------------------------------------------------------------------------
// Kernel 1: one wave <-> 16 frequencies. Y (B-layout, complex as 2 planes),
// iterate Y <- Z∘(VG·Y), S += Y; finally H = C∘S.
// --------------------------------------------------------------------------
__global__ void __launch_bounds__(256) fdn_solve_kernel(
    const float* __restrict__ theta, const float* __restrict__ Bv,
    const float* __restrict__ Cv, const float* __restrict__ VG,
    float* __restrict__ out) {
    const int tid  = threadIdx.x;
    const int lane = tid & 31;
    const int h    = lane >> 4;   // wave half
    const int c    = lane & 15;   // frequency column within batch
    const int wave = (blockIdx.x * blockDim.x + tid) >> 5;
    const int f0   = wave * 16;

    const float th = theta[f0 + c];

    // VG in A-operand layout: chunk kk covers K = 4kk..4kk+3.
    // lanes 0-15: K = 4kk+{0,1}; lanes 16-31: K = 4kk+{2,3}; lane&15 = M row.
    v2f Avg[4];
#pragma unroll
    for (int kk = 0; kk < 4; ++kk) {
        Avg[kk].x = VG[c * 16 + 4 * kk + 2 * h + 0];
        Avg[kk].y = VG[c * 16 + 4 * kk + 2 * h + 1];
    }

    // Per-lane state in B-layout: reg (kk,j) holds row n = 4kk+j+2h, col c.
    float Zre[4][2], Zim[4][2], Yre[4][2], Yim[4][2];
    float Sre[4][2], Sim[4][2], Cc[4][2];
#pragma unroll
    for (int kk = 0; kk < 4; ++kk)
#pragma unroll
        for (int j = 0; j < 2; ++j) {
            const int n = 4 * kk + j + 2 * h;
            // z^{-m_n} = exp(-i m_n θ); double-precision range reduction
            double ang = fmod(-(double)c_delays[n] * (double)th,
                              6.283185307179586);
            float si, co;
            sincosf((float)ang, &si, &co);
            Zre[kk][j] = co;
            Zim[kk][j] = si;
            const float bn = Bv[n];
            Cc[kk][j]  = Cv[n];
            Yre[kk][j] = co * bn;          // Y0 = D^{-1} B
            Yim[kk][j] = si * bn;
            Sre[kk][j] = Yre[kk][j];
            Sim[kk][j] = Yim[kk][j];
        }

    float TreB[4][2], TimB[4][2];
    for (int it = 0; it < NITER; ++it) {
        v8f Tre = {};
        v8f Tim = {};
#pragma unroll
        for (int kk = 0; kk < 4; ++kk) {
            v2f br, bi;
            br.x = Yre[kk][0]; br.y = Yre[kk][1];
            bi.x = Yim[kk][0]; bi.y = Yim[kk][1];
            // T += VG_chunk × Y_chunk  (real matmul, re/im planes interleaved)
            Tre = __builtin_amdgcn_wmma_f32_16x16x4_f32(
                false, Avg[kk], false, br, (short)0, Tre, false, false);
            Tim = __builtin_amdgcn_wmma_f32_16x16x4_f32(
                false, Avg[kk], false, bi, (short)0, Tim, false, false);
        }
        d_to_b(Tre, TreB, h);
        d_to_b(Tim, TimB, h);
#pragma unroll
        for (int kk = 0; kk < 4; ++kk)
#pragma unroll
            for (int j = 0; j < 2; ++j) {
                const float yr = Zre[kk][j] * TreB[kk][j] - Zim[kk][j] * TimB[kk][j];
                const float yi = Zre[kk][j] * TimB[kk][j] + Zim[kk][j] * TreB[kk][j];
                Yre[kk][j] = yr;
                Yim[kk][j] = yi;
                Sre[kk][j] += yr;
                Sim[kk][j] += yi;
            }
    }

    // H[f, n] = C_n * S[n, f]; complex64 stored as interleaved float pairs
    float2* out2 = reinterpret_cast<float2*>(out);
#pragma unroll
    for (int kk = 0; kk < 4; ++kk)
#pragma unroll
        for (int j = 0; j < 2; ++j) {
            const int n = 4 * kk + j + 2 * h;
            const int f = f0 + c;
            out2[f * FDN_N + n] =
                make_float2(Cc[kk][j] * Sre[kk][j], Cc[kk][j] * Sim[kk][j]);
        }
}

extern "C" void kernel_launch(void* const* d_in, const int* in_sizes, int n_in,
                              void* d_out, int out_size, void* d_ws,
                              size_t ws_size, hipStream_t stream) {
    const float* theta = (const float*)d_in[0];  // [131072]
    const float* A     = (const float*)d_in[1];  // [16,16]
    const float* Bv    = (const float*)d_in[2];  // [16]
    const float* Cv    = (const float*)d_in[3];  // [16]
    float* VG          = (float*)d_ws;           // 256 floats scratch
    float* out         = (float*)d_out;          // 131072*16 complex64

    fdn_expm_kernel<<<1, 256, 0, stream>>>(A, VG);

    const int waves_total = FDN_NUM / 16;              // 8192 waves
    const int block       = 256;                       // 8 waves per block
    const int grid        = waves_total / (block / 32);// 1024 blocks
    fdn_solve_kernel<<<grid, block, 0, stream>>>(theta, Bv, Cv, VG, out);
}